// SpatialPyramidPooling_50457275793429
// MI455X (gfx1250) — compile-verified
//
#include <hip/hip_runtime.h>
#include <hip/hip_bf16.h>

typedef __attribute__((ext_vector_type(2))) float v2f;
typedef __attribute__((ext_vector_type(8))) float v8f;

// ---------------------------------------------------------------------------
// Kernel 1: streaming window partial sums (bandwidth-critical, ~268 MB read).
// Each block: one (batch, 16x16 window, 4-row group). 256 threads, each owns
// 2 consecutive channels (float2) -> per spatial position the block reads a
// contiguous 2 KB line (perfectly coalesced). 64 positions per thread,
// register accumulation, no LDS. ws[b][k][c], k = win*4 + rowgroup (64 rows).
// ---------------------------------------------------------------------------
__global__ void spp_winsum_kernel(const float* __restrict__ in,
                                  float* __restrict__ ws) {
    const int tid = threadIdx.x;          // 0..255 -> channel pair
    const int rg  = blockIdx.x;           // 0..3   row group (4 rows)
    const int win = blockIdx.y;           // 0..15  16x16 window
    const int b   = blockIdx.z;           // 0..31

    const int wi = win >> 2, wj = win & 3;
    const int h0 = wi * 16 + rg * 4;
    const int w0 = wj * 16;

    const float2* inp = (const float2*)in;

    float2 acc0 = make_float2(0.f, 0.f);
    float2 acc1 = make_float2(0.f, 0.f);

#pragma unroll
    for (int r = 0; r < 4; ++r) {
        const int rowbase = ((b * 64 + h0 + r) * 64 + w0) * 256 + tid;
#pragma unroll
        for (int cc = 0; cc < 16; cc += 2) {
            float2 v0 = inp[rowbase + cc * 256];
            float2 v1 = inp[rowbase + (cc + 1) * 256];
            acc0.x += v0.x; acc0.y += v0.y;
            acc1.x += v1.x; acc1.y += v1.y;
        }
    }

    float2 s;
    s.x = acc0.x + acc1.x;
    s.y = acc0.y + acc1.y;
    ((float2*)ws)[(b * 64 + win * 4 + rg) * 256 + tid] = s;
}

// ---------------------------------------------------------------------------
// Combine matrix coefficient: out row m (0..20) vs window win (0..15).
//   m==0          : s=1 bin,  1/4096 of every window
//   m in 1..4     : s=2 bin (i,j) = ((m-1)>>1, (m-1)&1), 1/1024 of 4 windows
//   m in 5..20    : s=4 bin == window m-5, 1/256
// Branchless (v_cndmask), so EXEC stays all-ones around the WMMAs.
// ---------------------------------------------------------------------------
__device__ __forceinline__ float spp_mval(int m, int win) {
    const int wi = win >> 2, wj = win & 3;
    float r = (m == 0) ? (1.0f / 4096.0f) : 0.0f;
    const int t = m - 1;
    const bool s2 = (m >= 1) & (m <= 4) &
                    ((wi >> 1) == (t >> 1)) & ((wj >> 1) == (t & 1));
    r += s2 ? (1.0f / 1024.0f) : 0.0f;
    const bool s4 = (m >= 5) & (m <= 20) & ((m - 5) == win);
    r += s4 ? (1.0f / 256.0f) : 0.0f;
    return r;
}

// ---------------------------------------------------------------------------
// Kernel 2: out[b, m*512 + c] = sum_k M[m,k] * W[b,k,c]  via
// V_WMMA_F32_16X16X4_F32. One wave per (batch, 16-channel group).
// K = 64 partial-sum rows -> 16 k-steps; two M tiles (rows 0-15, 16-20).
// A layout (16x4 f32): lanes 0-15 K=0/1 in a.x/a.y, lanes 16-31 K=2/3 —
// coefficient is identical for all 4 K of a step (same window), so A is a
// broadcast. B layout (4x16): v0 = K0|K2 halves, v1 = K1|K3 halves.
// D layout: VGPR r holds M = r + 8*(lane>=16), N = lane&15.
// ---------------------------------------------------------------------------
__global__ void spp_combine_wmma_kernel(const float* __restrict__ ws,
                                        float* __restrict__ out) {
    const int lane  = threadIdx.x;        // 0..31 (wave32)
    const int c0    = blockIdx.x * 16;    // channel group
    const int b     = blockIdx.y;         // batch
    const int n     = lane & 15;          // column within tile
    const int khalf = lane >> 4;          // 0: K=0,1  1: K=2,3

    const float* wsb = ws + (size_t)b * 64 * 512 + c0 + n;

    v8f acc0 = {};   // output rows 0..15
    v8f acc1 = {};   // output rows 16..20 (21..31 padded with zero coeffs)

#pragma unroll
    for (int kk = 0; kk < 16; ++kk) {
        const int k0 = kk * 4 + 2 * khalf;
        v2f bmat;
        bmat.x = wsb[(size_t)k0 * 512];
        bmat.y = wsb[(size_t)(k0 + 1) * 512];

        const float a0 = spp_mval(n, kk);        // rows 0..15
        const float a1 = spp_mval(16 + n, kk);   // rows 16..31 (>=21 -> 0)
        v2f av0; av0.x = a0; av0.y = a0;
        v2f av1; av1.x = a1; av1.y = a1;

        acc0 = __builtin_amdgcn_wmma_f32_16x16x4_f32(
            false, av0, false, bmat, (short)0, acc0, false, false);
        acc1 = __builtin_amdgcn_wmma_f32_16x16x4_f32(
            false, av1, false, bmat, (short)0, acc1, false, false);
    }

    float* outb = out + (size_t)b * 10752 + c0 + n;  // 21*512 = 10752 per batch
#pragma unroll
    for (int r = 0; r < 8; ++r) {
        const int m0 = r + 8 * khalf;                // 0..15: always valid
        outb[(size_t)m0 * 512] = acc0[r];
    }
#pragma unroll
    for (int r = 0; r < 8; ++r) {
        const int m1 = 16 + r + 8 * khalf;           // valid only for 16..20
        if (m1 < 21) outb[(size_t)m1 * 512] = acc1[r];
    }
}

// ---------------------------------------------------------------------------
// Launch: kernel1 (2048 blocks x 256 thr = 16384 waves, HBM-saturating
// streaming reduction), then kernel2 (1024 single-wave blocks, 32 WMMAs each).
// Workspace use: 32*64*512*4 = 4 MB.
// ---------------------------------------------------------------------------
extern "C" void kernel_launch(void* const* d_in, const int* in_sizes, int n_in,
                              void* d_out, int out_size, void* d_ws, size_t ws_size,
                              hipStream_t stream) {
    const float* in = (const float*)d_in[0];
    float* out = (float*)d_out;
    float* ws  = (float*)d_ws;

    dim3 g1(4, 16, 32);          // (rowgroup, window, batch)
    spp_winsum_kernel<<<g1, dim3(256), 0, stream>>>(in, ws);

    dim3 g2(32, 32);             // (channel group of 16, batch)
    spp_combine_wmma_kernel<<<g2, dim3(32), 0, stream>>>(ws, out);
}